// VQEmbedding_59442347377481
// MI455X (gfx1250) — compile-verified
//
#include <hip/hip_runtime.h>

typedef __attribute__((ext_vector_type(2))) float v2f;
typedef __attribute__((ext_vector_type(8))) float v8f;
typedef __attribute__((ext_vector_type(4))) unsigned int u32x4;
typedef __attribute__((ext_vector_type(8))) unsigned int u32x8;

#define KCODES 1024
#define DDIM   256
#define HW     4096                    // 64*64 spatial per batch image
#define CHUNK_FLOATS (16 * 260)        // padded 16x256 chunk (260-float stride)
#define CHUNK_BYTES  (CHUNK_FLOATS * 4)

// ---------------------------------------------------------------------------
// Kernel 1: e_sq[c] = sum_d emb[c][d]^2
// ---------------------------------------------------------------------------
__global__ void vq_esq_kernel(const float* __restrict__ emb,
                              float* __restrict__ esq) {
    const int c = blockIdx.x * 128 + threadIdx.x;   // 8 blocks * 128 = 1024
    const float4* e = (const float4*)(emb + c * DDIM);
    float s = 0.f;
    #pragma unroll
    for (int i = 0; i < DDIM / 4; ++i) {
        float4 v = e[i];
        s += v.x * v.x + v.y * v.y + v.z * v.z + v.w * v.w;
    }
    esq[c] = s;
}

// ---------------------------------------------------------------------------
// TDM: DMA one 16-code x 256-dim f32 slice of the codebook into LDS.
// Hardware padding (pad_interval = 256 DWORDs, pad_amount = 4 DWORDs)
// reproduces the 260-float row stride -> bank-conflict-free ds_load_b64.
// D# layout per CDNA5 ISA ch.8 (group0 128b + group1 256b, 2D tensor form).
// ---------------------------------------------------------------------------
__device__ __forceinline__ void tdm_stage(const float* gsrc, unsigned lds_addr) {
    unsigned long long ga = (unsigned long long)(uintptr_t)gsrc;
    u32x4 g0;
    g0[0] = 1u;                                   // count=1, user descriptor
    g0[1] = lds_addr;                             // LDS byte address
    g0[2] = (unsigned)ga;                         // global_addr[31:0]
    g0[3] = (unsigned)((ga >> 32) & 0x01FFFFFFu)  // global_addr[56:32]
          | 0x80000000u;                          // type=2 ("image")
    u32x8 g1;
    g1[0] = (2u << 16)        // data_size = 4 bytes
          | (1u << 20)        // pad_enable
          | (7u << 22)        // pad_interval: pad every 256 DWORDs
          | (3u << 25);       // pad_amount: 4 DWORDs
    g1[1] = 256u << 16;       // tensor_dim0 = 256 (bits 79:48)
    g1[2] = 16u  << 16;       // tensor_dim1 = 16  (bits 111:80)
    g1[3] = 256u << 16;       // tile_dim0   = 256 (bits 127:112)
    g1[4] = 16u;              // tile_dim1   = 16  (bits 143:128); tile_dim2=0
    g1[5] = 256u;             // tensor_dim0_stride = 256 (bits 207:160)
    g1[6] = 0u;
    g1[7] = 0u;
    asm volatile("tensor_load_to_lds %0, %1" :: "s"(g0), "s"(g1) : "memory");
}

// ---------------------------------------------------------------------------
// Kernel 2: per 64-pixel block, sweep all 1024 codes with f32 WMMA; running
// argmin of (||e||^2 - 2*x.e). TDM double-buffers the codebook slices.
// ---------------------------------------------------------------------------
__launch_bounds__(128)
__global__ void vq_argmin_kernel(const float* __restrict__ z,
                                 const float* __restrict__ emb,
                                 const float* __restrict__ esq,
                                 float* __restrict__ out) {
    __shared__ __align__(16) float sB[2][CHUNK_FLOATS];
    __shared__ float sE[KCODES];
    __shared__ float redV[64][16];
    __shared__ int   redI[64][16];

    const int tid   = threadIdx.x;
    const int wave  = tid >> 5;      // 0..3
    const int lane  = tid & 31;
    const int laneM = lane & 15;     // matrix row (A) / code within chunk (B,C)
    const int laneH = lane >> 4;     // half-wave selector (K pair / M upper half)

    const unsigned sB0 = (unsigned)(uintptr_t)&sB[0][0];  // LDS byte offset

    // kick off DMA of chunk 0 before touching anything else
    if (wave == 0) tdm_stage(emb, sB0);

    // stage ||e||^2 once
    for (int i = tid; i < KCODES; i += 128) sE[i] = esq[i];

    // ------- load A fragments: 16 pixels x 256 dims per wave, kept in VGPRs ---
    const int p  = blockIdx.x * 64 + wave * 16 + laneM;
    const int b  = p >> 12;
    const int hw = p & (HW - 1);
    // A[M=laneM][k] for k = 4g + 2*laneH + {0,1}  (ISA 16x4 f32 A layout)
    const float* zp = z + ((b * DDIM + 2 * laneH) * HW + hw);
    float a0[64], a1[64];
    #pragma unroll
    for (int g = 0; g < 64; ++g) {
        a0[g] = zp[(4 * g) * HW];
        a1[g] = zp[(4 * g + 1) * HW];
    }

    float best[8];
    int   bidx[8];
    #pragma unroll
    for (int r = 0; r < 8; ++r) { best[r] = 3.4e38f; bidx[r] = 0; }

    for (int kk = 0; kk < 64; ++kk) {
        if (wave == 0) __builtin_amdgcn_s_wait_tensorcnt(0);
        __syncthreads();   // chunk kk visible; buffer (kk+1)&1 free of readers
        if (wave == 0 && kk < 63)
            tdm_stage(emb + (kk + 1) * (16 * DDIM),
                      sB0 + (unsigned)(((kk + 1) & 1) * CHUNK_BYTES));

        // two independent accumulator chains for WMMA ILP
        v8f acc0 = {};
        v8f acc1 = {};
        const float* bp = &sB[kk & 1][laneM * 260 + 2 * laneH];
        #pragma unroll
        for (int g = 0; g < 64; g += 2) {
            v2f ae; ae.x = a0[g];     ae.y = a1[g];
            v2f be = *(const v2f*)(bp + 4 * g);
            acc0 = __builtin_amdgcn_wmma_f32_16x16x4_f32(
                false, ae, false, be, (short)0, acc0, false, false);
            v2f ao; ao.x = a0[g + 1]; ao.y = a1[g + 1];
            v2f bo = *(const v2f*)(bp + 4 * (g + 1));
            acc1 = __builtin_amdgcn_wmma_f32_16x16x4_f32(
                false, ao, false, bo, (short)0, acc1, false, false);
        }

        // C/D layout: N = laneM (fixed per lane), M = r + 8*laneH
        const int   code = kk * 16 + laneM;
        const float ev   = sE[code];
        #pragma unroll
        for (int r = 0; r < 8; ++r) {
            float d = ev - 2.0f * (acc0[r] + acc1[r]);
            if (d < best[r]) { best[r] = d; bidx[r] = code; }
        }
    }

    // ------- cross-lane argmin: 16 candidates (code residues) per pixel ------
    #pragma unroll
    for (int r = 0; r < 8; ++r) {
        int pl = wave * 16 + laneH * 8 + r;   // pixel-in-block
        redV[pl][laneM] = best[r];
        redI[pl][laneM] = bidx[r];
    }
    __syncthreads();
    if (tid < 64) {
        float bv = redV[tid][0];
        int   bi = redI[tid][0];
        #pragma unroll
        for (int s = 1; s < 16; ++s) {
            float v  = redV[tid][s];
            int   i2 = redI[tid][s];
            if (v < bv || (v == bv && i2 < bi)) { bv = v; bi = i2; }
        }
        out[blockIdx.x * 64 + tid] = (float)bi;
    }
}

// ---------------------------------------------------------------------------
extern "C" void kernel_launch(void* const* d_in, const int* in_sizes, int n_in,
                              void* d_out, int out_size, void* d_ws, size_t ws_size,
                              hipStream_t stream) {
    const float* z   = (const float*)d_in[0];   // [32,256,64,64] fp32
    const float* emb = (const float*)d_in[1];   // [1024,256]     fp32
    float* esq = (float*)d_ws;                  // 1024 floats scratch
    float* out = (float*)d_out;                 // 131072 indices (as float)

    vq_esq_kernel<<<8, 128, 0, stream>>>(emb, esq);
    vq_argmin_kernel<<<2048, 128, 0, stream>>>(z, emb, esq, out);
}